// DifferentiableSolver_63754494542222
// MI455X (gfx1250) — compile-verified
//
#include <hip/hip_runtime.h>

// ---------------------------------------------------------------------------
// Batched QP via ADMM, reformulated as x = Q v + w per iteration.
//   sigma = 2*L2_REG + RHO = 1.2,  RHO = 1,  JITTER = 1e-5
//   M  = A A^T + jit*I (16x16) ; Minv via wave-parallel Gauss-Jordan
//   Y  = Minv * A      (16x32) ; WMMA
//   Qf = A^T * Y       (32x32) ; WMMA
//   Q  = (I - Qf)/sigma ; w = Y^T b
// Iterate 100x: v = (z-u)-c ; x = Q v + w ; z = clip(x+u,lb,ub) ; u = x+u-z
// One wave32 per sample. WMMA f32 16x16x4 used for all setup GEMMs.
// Gauss-Jordan pivot broadcasts use v_readlane (uniform lane index) instead
// of ds_bpermute to keep the DS pipe free and avoid dscnt waits.
// ---------------------------------------------------------------------------

typedef float v2f __attribute__((ext_vector_type(2)));
typedef float v8f __attribute__((ext_vector_type(8)));

__device__ __forceinline__ v8f wmma_f32_16x16x4(v2f a, v2f b, v8f c) {
  // (neg_a, A, neg_b, B, c_mod, C, reuse_a, reuse_b)
  return __builtin_amdgcn_wmma_f32_16x16x4_f32(false, a, false, b, (short)0, c,
                                               false, false);
}

// Broadcast lane `l` (compile-time uniform) -> all lanes via v_readlane_b32.
__device__ __forceinline__ float bcast_lane(float v, int l) {
  return __int_as_float(__builtin_amdgcn_readlane(__float_as_int(v), l));
}

#define WAVE_LDS_FLOATS 1536  // 6 KB per wave
#define WAVES_PER_BLOCK 4

__global__ void __launch_bounds__(128) admm_qp_kernel(
    const float* __restrict__ Ag, const float* __restrict__ bg,
    const float* __restrict__ cg, const float* __restrict__ lbg,
    const float* __restrict__ ubg, float* __restrict__ outg, int nS) {
  __shared__ __align__(16) float smem[WAVES_PER_BLOCK * WAVE_LDS_FLOATS];

  const int lane = threadIdx.x & 31;
  const int wid = threadIdx.x >> 5;
  const int s = blockIdx.x * WAVES_PER_BLOCK + wid;
  if (s >= nS) return;  // wave-uniform

  // Per-wave LDS regions (floats). Q (stride 40, rows 0..31 -> 1280 floats)
  // overlays A/Y/G after they are consumed; V overlays Minv during iterations.
  float* W = smem + wid * WAVE_LDS_FLOATS;
  float* Alds = W;          // 16x32 row-major (512)
  float* Ylds = W + 512;    // 16x32 row-major (512)
  float* Glds = W + 1024;   // 16x16 row-major (256)
  float* Milds = W + 1280;  // 16x16 row-major (256)
  float* Qlds = W;          // 32 rows, stride 40 (1280)
  float* Vlds = W + 1280;   // 32 floats broadcast slot

  // ---- stage A (row-major 16x32) into LDS -------------------------------
  {
    const float4* Agv = (const float4*)(Ag + (size_t)s * 512);
    float4* Aldv = (float4*)Alds;
#pragma unroll
    for (int i = 0; i < 4; ++i) Aldv[lane + 32 * i] = Agv[lane + 32 * i];
  }

  const int m16 = lane & 15;
  const int hi = (lane < 16) ? 0 : 2;  // K-offset within 16x4 fragment
  const int rb = (lane < 16) ? 0 : 8;  // M-offset within C/D tile

  // ---- G = A A^T + jitter*I  (8x WMMA f32 16x16x4, K=32) ----------------
  {
    v8f g = {};
#pragma unroll
    for (int k = 0; k < 8; ++k) {
      // A-fragment of A[:,4k:4k+4]; same regs serve as B-fragment of its
      // transpose (symmetric K-split lane layout).
      v2f af = *(const v2f*)&Alds[m16 * 32 + 4 * k + hi];
      g = wmma_f32_16x16x4(af, af, g);
    }
#pragma unroll
    for (int r = 0; r < 8; ++r) {
      int row = rb + r;
      Glds[row * 16 + m16] = g[r] + ((row == m16) ? 1e-5f : 0.0f);
    }
  }

  // ---- Gauss-Jordan: lane j holds column j of [G | I] -------------------
  float wc[16];
#pragma unroll
  for (int p = 0; p < 16; ++p) {
    float gv = Glds[p * 16 + m16];
    wc[p] = (lane < 16) ? gv : ((p == lane - 16) ? 1.0f : 0.0f);
  }
#pragma unroll
  for (int p = 0; p < 16; ++p) {
    float piv = bcast_lane(wc[p], p);
    float ip = 1.0f / piv;
    wc[p] *= ip;
#pragma unroll
    for (int r = 0; r < 16; ++r) {
      if (r == p) continue;
      float f = bcast_lane(wc[r], p);
      wc[r] = fmaf(-f, wc[p], wc[r]);
    }
  }
  if (lane >= 16) {
#pragma unroll
    for (int p = 0; p < 16; ++p) Milds[p * 16 + (lane - 16)] = wc[p];
  }

  // ---- Y = Minv * A  (2 tiles x 4 WMMA, K=16) ---------------------------
  {
    v2f mf[4];
#pragma unroll
    for (int k = 0; k < 4; ++k)
      mf[k] = *(const v2f*)&Milds[m16 * 16 + 4 * k + hi];
#pragma unroll
    for (int t = 0; t < 2; ++t) {
      v8f y = {};
#pragma unroll
      for (int k = 0; k < 4; ++k) {
        int r0 = 4 * k + hi;
        v2f bf;
        bf.x = Alds[r0 * 32 + 16 * t + m16];
        bf.y = Alds[(r0 + 1) * 32 + 16 * t + m16];
        y = wmma_f32_16x16x4(mf[k], bf, y);
      }
#pragma unroll
      for (int r = 0; r < 8; ++r) Ylds[(rb + r) * 32 + 16 * t + m16] = y[r];
    }
  }

  // ---- w = Y^T b (lane holds w[lane]); b loads are wave-uniform ---------
  float w_ = 0.0f;
  {
    const float* bptr = bg + (size_t)s * 16;
#pragma unroll
    for (int m = 0; m < 16; ++m) w_ = fmaf(Ylds[m * 32 + lane], bptr[m], w_);
  }

  // ---- Qfull = A^T * Y (4 tiles x 4 WMMA, K=16) -------------------------
  v8f qt[2][2];
#pragma unroll
  for (int i = 0; i < 2; ++i) {
#pragma unroll
    for (int jt = 0; jt < 2; ++jt) {
      v8f q = {};
#pragma unroll
      for (int k = 0; k < 4; ++k) {
        int r0 = 4 * k + hi;
        v2f at, bt;
        at.x = Alds[r0 * 32 + 16 * i + m16];  // A^T fragment
        at.y = Alds[(r0 + 1) * 32 + 16 * i + m16];
        bt.x = Ylds[r0 * 32 + 16 * jt + m16];  // Y fragment
        bt.y = Ylds[(r0 + 1) * 32 + 16 * jt + m16];
        q = wmma_f32_16x16x4(at, bt, q);
      }
      qt[i][jt] = q;
    }
  }

  // ---- Q = (I - Qfull)/sigma, written over A/Y region (stride 40) -------
  const float invsig = 1.0f / 1.2f;
#pragma unroll
  for (int i = 0; i < 2; ++i) {
#pragma unroll
    for (int jt = 0; jt < 2; ++jt) {
#pragma unroll
      for (int r = 0; r < 8; ++r) {
        int grow = 16 * i + rb + r;
        int gcol = 16 * jt + m16;
        float idv = (grow == gcol) ? 1.0f : 0.0f;
        Qlds[grow * 40 + gcol] = (idv - qt[i][jt][r]) * invsig;
      }
    }
  }

  // ---- lane i pulls row i of Q into registers ---------------------------
  float q[32];
  {
    const float4* qrow = (const float4*)&Qlds[lane * 40];
#pragma unroll
    for (int jj = 0; jj < 8; ++jj) {
      float4 t4 = qrow[jj];
      q[4 * jj + 0] = t4.x;
      q[4 * jj + 1] = t4.y;
      q[4 * jj + 2] = t4.z;
      q[4 * jj + 3] = t4.w;
    }
  }

  // ---- ADMM state (lane k owns component k) -----------------------------
  const float c_ = cg[(size_t)s * 32 + lane];
  const float lb_ = lbg[(size_t)s * 32 + lane];
  const float ub_ = ubg[(size_t)s * 32 + lane];
  float x_ = 0.0f;
  float z_ = fminf(fmaxf(0.0f, lb_), ub_);
  float u_ = 0.0f;

  const float4* vv = (const float4*)Vlds;
  for (int it = 0; it < 100; ++it) {
    float v = (z_ - u_) - c_;  // RHO = 1
    Vlds[lane] = v;            // same-wave DS is in-order: no barrier needed
    float4 vb[8];
#pragma unroll
    for (int jj = 0; jj < 8; ++jj) vb[jj] = vv[jj];  // broadcast reads
    float a0 = w_, a1 = 0.0f, a2 = 0.0f, a3 = 0.0f;
#pragma unroll
    for (int jj = 0; jj < 8; ++jj) {
      a0 = fmaf(q[4 * jj + 0], vb[jj].x, a0);
      a1 = fmaf(q[4 * jj + 1], vb[jj].y, a1);
      a2 = fmaf(q[4 * jj + 2], vb[jj].z, a2);
      a3 = fmaf(q[4 * jj + 3], vb[jj].w, a3);
    }
    x_ = (a0 + a1) + (a2 + a3);
    float t = x_ + u_;
    z_ = fminf(fmaxf(t, lb_), ub_);
    u_ = t - z_;
  }

  outg[(size_t)s * 32 + lane] = x_;
}

extern "C" void kernel_launch(void* const* d_in, const int* in_sizes, int n_in,
                              void* d_out, int out_size, void* d_ws,
                              size_t ws_size, hipStream_t stream) {
  (void)n_in; (void)out_size; (void)d_ws; (void)ws_size;
  const float* A = (const float*)d_in[0];
  const float* b = (const float*)d_in[1];
  const float* c = (const float*)d_in[2];
  const float* lb = (const float*)d_in[3];
  const float* ub = (const float*)d_in[4];
  float* out = (float*)d_out;
  const int nS = in_sizes[1] / 16;  // b is (B,16)
  const int blocks = (nS + WAVES_PER_BLOCK - 1) / WAVES_PER_BLOCK;
  admm_qp_kernel<<<blocks, 32 * WAVES_PER_BLOCK, 0, stream>>>(A, b, c, lb, ub,
                                                              out, nS);
}